// MultiBoxLossOnline_67980742361440
// MI455X (gfx1250) — compile-verified
//
#include <hip/hip_runtime.h>
#include <stdint.h>

#define BB 16
#define AA 32768
#define CC 81
#define NEGR 3

// ---- kernel1 configuration ----
#define K1_BLOCKS 2048
#define K1_THREADS 64          // 2 waves per block (wave32)
#define TILES_PER_WAVE 4       // 2048 blocks * 2 waves * 4 tiles = 16384 tiles = B*A/32
#define TILE_F32 (32 * CC)     // 2592 floats
#define TILE_BYTES (TILE_F32 * 4)  // 10368 bytes, multiple of 16

// ---- kernel2 configuration ----
#define K2_THREADS 1024        // 32 waves; row staged in 128 KB LDS
#define NHIST 16               // privatized histograms (one per 64 threads)
#define HSTRIDE 257            // pad: hist h, bin b -> LDS bank (h+b) mod 64

// ---- workspace layout (bytes) ----
#define WS_CENEG   0                       // float[B*A]  = 2 MB
#define WS_NUMPOS  (BB * AA * 4)           // int[16]
#define WS_BLKLOC  (WS_NUMPOS + 64)        // float[K1_BLOCKS]
#define WS_BLKCLS  (WS_BLKLOC + K1_BLOCKS * 4)
#define WS_NEGSUM  (WS_BLKCLS + K1_BLOCKS * 4)  // float[16]

__device__ __forceinline__ float smooth_l1(float d) {
    float ad = fabsf(d);
    return ad < 1.0f ? 0.5f * d * d : ad - 0.5f;
}

__global__ void k0_init(int* num_pos) {
    if (threadIdx.x < BB) num_pos[threadIdx.x] = 0;
}

// ------------------------------------------------------------------
// Kernel 1: CE per anchor (logsumexp over 81 classes from LDS tile,
// staged via async global->LDS DMA, double buffered), masked
// smooth-L1, deterministic per-block partial sums.
// ------------------------------------------------------------------
__global__ __launch_bounds__(K1_THREADS)
void k1_ce(const float* __restrict__ loc_p, const float* __restrict__ cls_p,
           const float* __restrict__ loc_t, const int* __restrict__ cls_t,
           const unsigned char* __restrict__ ignore,
           float* __restrict__ ce_neg, int* __restrict__ num_pos,
           float* __restrict__ blk_loc, float* __restrict__ blk_cls)
{
    // 2 waves * 2 buffers * 2592 floats = 41472 B, + small reduction arrays
    __shared__ __align__(16) float smem[2 * 2 * TILE_F32];
    __shared__ float red_loc[K1_THREADS];
    __shared__ float red_cls[K1_THREADS];
    __shared__ int   red_np[K1_THREADS];

    const int tid  = threadIdx.x;
    const int lane = tid & 31;
    const int wv   = tid >> 5;                      // 0..1
    const int gw   = blockIdx.x * 2 + wv;           // global wave id
    const int tile0 = gw * TILES_PER_WAVE;

    const uint32_t lds0 = (uint32_t)(uintptr_t)&smem[0];   // LDS byte offset of smem
    const uint32_t slab = lds0 + (uint32_t)(wv * 2 * TILE_BYTES);
    const uint64_t cls_base = (uint64_t)(uintptr_t)cls_p;

    float acc_loc = 0.0f, acc_cls = 0.0f;
    int   acc_np  = 0;

    // Issue one tile's worth of async global->LDS copies: 20 x b128 + 1 x b32
    auto issue = [&](int buf, int tile) {
        uint64_t g = cls_base + (uint64_t)tile * TILE_BYTES + (uint32_t)(lane * 16);
        uint32_t l = slab + (uint32_t)(buf * TILE_BYTES) + (uint32_t)(lane * 16);
#pragma unroll
        for (int j = 0; j < 20; ++j) {
            asm volatile("global_load_async_to_lds_b128 %0, %1, off"
                         :: "v"(l), "v"(g) : "memory");
            l += 512; g += 512;
        }
        uint64_t g32 = cls_base + (uint64_t)tile * TILE_BYTES + 10240u + (uint32_t)(lane * 4);
        uint32_t l32 = slab + (uint32_t)(buf * TILE_BYTES) + 10240u + (uint32_t)(lane * 4);
        asm volatile("global_load_async_to_lds_b32 %0, %1, off"
                     :: "v"(l32), "v"(g32) : "memory");
    };

    issue(0, tile0);
    for (int i = 0; i < TILES_PER_WAVE; ++i) {
        if (i + 1 < TILES_PER_WAVE) {
            issue((i + 1) & 1, tile0 + i + 1);
            // 21 newer copies in flight; previous tile's 21 are done (in-order)
            asm volatile("s_wait_asynccnt 21" ::: "memory");
        } else {
            asm volatile("s_wait_asynccnt 0" ::: "memory");
        }

        const int tile = tile0 + i;
        const int g = tile * 32 + lane;                 // flat anchor index in [0, B*A)
        const float* x = &smem[(wv * 2 + (i & 1)) * TILE_F32 + lane * CC];

        // logsumexp over 81 classes (bank-conflict-free: stride 81 -> 17*lane mod 64)
        float m = x[0];
        for (int c = 1; c < CC; ++c) m = fmaxf(m, x[c]);
        float s = 0.0f;
        for (int c = 0; c < CC; ++c) s += __expf(x[c] - m);
        const int t = cls_t[g];
        const float ce = __logf(s) + m - x[t];

        const bool pos = (t != 0);
        const bool neg = (!pos) && (ignore[g] == 0);
        ce_neg[g] = neg ? ce : 0.0f;

        if (pos) {
            acc_np += 1;
            acc_cls += ce;
            const float4 p  = ((const float4*)loc_p)[g];
            const float4 tt = ((const float4*)loc_t)[g];
            acc_loc += smooth_l1(p.x - tt.x / 0.1f);
            acc_loc += smooth_l1(p.y - tt.y / 0.1f);
            acc_loc += smooth_l1(p.z - tt.z / 0.2f);
            acc_loc += smooth_l1(p.w - tt.w / 0.2f);
        }
    }

    // deterministic block tree reduction
    red_loc[tid] = acc_loc; red_cls[tid] = acc_cls; red_np[tid] = acc_np;
    __syncthreads();
    for (int s2 = K1_THREADS / 2; s2 > 0; s2 >>= 1) {
        if (tid < s2) {
            red_loc[tid] += red_loc[tid + s2];
            red_cls[tid] += red_cls[tid + s2];
            red_np[tid]  += red_np[tid + s2];
        }
        __syncthreads();
    }
    if (tid == 0) {
        blk_loc[blockIdx.x] = red_loc[0];
        blk_cls[blockIdx.x] = red_cls[0];
        const int b = blockIdx.x >> 7;   // 256 anchors per block, 128 blocks per batch row
        atomicAdd(&num_pos[b], red_np[0]);
    }
}

// ------------------------------------------------------------------
// Kernel 2: exact top-k sum per batch row. One global sweep stages
// the 128 KB row into LDS; 4-pass radix select on nonnegative float
// bit patterns with bank-spread privatized histograms and a parallel
// suffix-scan bin selection. Exact tie handling matches sort order.
// ------------------------------------------------------------------
__global__ __launch_bounds__(K2_THREADS)
void k2_topk(const float* __restrict__ ce_neg, const int* __restrict__ num_pos,
             float* __restrict__ neg_sum)
{
    __shared__ float vals[AA];              // 131072 B row stage
    __shared__ int hist[NHIST * HSTRIDE];   // 16448 B privatized histograms
    __shared__ int redh[256];               // combined histogram / scan ping
    __shared__ int sfx[256];                // scan pong
    __shared__ unsigned s_prefix;
    __shared__ int s_rem;
    __shared__ float rsum[K2_THREADS];
    __shared__ int   rcnt[K2_THREADS];

    const int b = blockIdx.x;
    const int tid = threadIdx.x;
    const float* row = ce_neg + (size_t)b * AA;

    int k0 = num_pos[b] * NEGR;
    if (k0 > AA) k0 = AA;
    if (k0 <= 0) { if (tid == 0) neg_sum[b] = 0.0f; return; }   // uniform per block

    // stage row into LDS (single coalesced global sweep)
    for (int idx = tid; idx < AA; idx += K2_THREADS) vals[idx] = row[idx];
    if (tid == 0) { s_prefix = 0u; s_rem = k0; }

    int* const myh = &hist[(tid >> 6) * HSTRIDE];   // 64 threads share one histogram

    for (int pass = 3; pass >= 0; --pass) {
        const int shift = pass * 8;
        for (int i = tid; i < NHIST * HSTRIDE; i += K2_THREADS) hist[i] = 0;
        __syncthreads();   // also covers vals staging / s_prefix init on first pass

        const unsigned prefix = s_prefix;
        const int rem = s_rem;
        const unsigned maskHigh = (pass == 3) ? 0u : (0xFFFFFFFFu << (shift + 8));
        for (int idx = tid; idx < AA; idx += K2_THREADS) {
            const unsigned u = __float_as_uint(vals[idx]);
            if ((u & maskHigh) == prefix)
                atomicAdd(&myh[(u >> shift) & 255], 1);
        }
        __syncthreads();

        // combine the NHIST private histograms
        if (tid < 256) {
            int s = 0;
            for (int h = 0; h < NHIST; ++h) s += hist[h * HSTRIDE + tid];
            redh[tid] = s;
        }
        __syncthreads();
        const int hbin = (tid < 256) ? redh[tid] : 0;

        // inclusive suffix scan: a[t] = sum_{j >= t} redh[j]  (8 ping-pong steps)
        int* a = redh; int* p2 = sfx;
        for (int d = 1; d < 256; d <<= 1) {
            if (tid < 256) p2[tid] = a[tid] + ((tid + d < 256) ? a[tid + d] : 0);
            __syncthreads();
            int* t2 = a; a = p2; p2 = t2;
        }   // after 8 swaps a == redh, holding inclusive suffix sums

        // unique selected bin: above < rem <= incl
        if (tid < 256) {
            const int incl = a[tid];
            const int above = incl - hbin;
            if (above < rem && rem <= incl) {
                s_prefix = prefix | ((unsigned)tid << shift);
                s_rem = rem - above;
            }
        }
        __syncthreads();
    }

    const unsigned T = s_prefix;          // bit pattern of the k0-th largest value
    const float Tf = __uint_as_float(T);
    float sum = 0.0f; int cnt = 0;
    for (int idx = tid; idx < AA; idx += K2_THREADS) {
        const float v = vals[idx];
        if (__float_as_uint(v) > T) { sum += v; ++cnt; }
    }
    rsum[tid] = sum; rcnt[tid] = cnt;
    __syncthreads();
    for (int s2 = K2_THREADS / 2; s2 > 0; s2 >>= 1) {
        if (tid < s2) { rsum[tid] += rsum[tid + s2]; rcnt[tid] += rcnt[tid + s2]; }
        __syncthreads();
    }
    if (tid == 0) neg_sum[b] = rsum[0] + Tf * (float)(k0 - rcnt[0]);
}

// ------------------------------------------------------------------
// Kernel 3: final deterministic reduction + loss combination.
// ------------------------------------------------------------------
__global__ __launch_bounds__(256)
void k3_final(const float* __restrict__ blk_loc, const float* __restrict__ blk_cls,
              const int* __restrict__ num_pos, const float* __restrict__ neg_sum,
              float* __restrict__ out)
{
    __shared__ float rl[256], rc[256];
    const int tid = threadIdx.x;
    float l = 0.0f, c = 0.0f;
    for (int i = tid; i < K1_BLOCKS; i += 256) { l += blk_loc[i]; c += blk_cls[i]; }
    rl[tid] = l; rc[tid] = c;
    __syncthreads();
    for (int s = 128; s > 0; s >>= 1) {
        if (tid < s) { rl[tid] += rl[tid + s]; rc[tid] += rc[tid + s]; }
        __syncthreads();
    }
    if (tid == 0) {
        int np = 0; float ns = 0.0f;
        for (int i = 0; i < BB; ++i) { np += num_pos[i]; ns += neg_sum[i]; }
        const float tp = (float)(np > 1 ? np : 1);
        out[0] = (rc[0] + ns) / tp + rl[0] / tp;
    }
}

extern "C" void kernel_launch(void* const* d_in, const int* in_sizes, int n_in,
                              void* d_out, int out_size, void* d_ws, size_t ws_size,
                              hipStream_t stream) {
    (void)in_sizes; (void)n_in; (void)out_size; (void)ws_size;
    const float* loc_p = (const float*)d_in[0];
    const float* cls_p = (const float*)d_in[1];
    const float* loc_t = (const float*)d_in[2];
    const int*   cls_t = (const int*)d_in[3];
    const unsigned char* ignore = (const unsigned char*)d_in[4];

    char* ws = (char*)d_ws;
    float* ce      = (float*)(ws + WS_CENEG);
    int*   num_pos = (int*)  (ws + WS_NUMPOS);
    float* blk_loc = (float*)(ws + WS_BLKLOC);
    float* blk_cls = (float*)(ws + WS_BLKCLS);
    float* neg_sum = (float*)(ws + WS_NEGSUM);

    k0_init<<<1, 64, 0, stream>>>(num_pos);
    k1_ce<<<K1_BLOCKS, K1_THREADS, 0, stream>>>(loc_p, cls_p, loc_t, cls_t, ignore,
                                                ce, num_pos, blk_loc, blk_cls);
    k2_topk<<<BB, K2_THREADS, 0, stream>>>(ce, num_pos, neg_sum);
    k3_final<<<1, 256, 0, stream>>>(blk_loc, blk_cls, num_pos, neg_sum, (float*)d_out);
}